// Bin_Mean_Shift_53017076302473
// MI455X (gfx1250) — compile-verified
//
#include <hip/hip_runtime.h>
#include <math.h>

// ---------------- problem constants ----------------
#define NPTS    131072
#define NSAMP   1000
#define NSEED   15625       // 5^6
#define NSEEDP  16384       // padded
#define SMAX    2048
#define KMAX    256
#define BW      0.15f
#define BW2     (BW*BW)
#define EPSP    1e-6f
#define INVH    (-0.5f/(BW*BW))

typedef __attribute__((ext_vector_type(2))) float v2f;
typedef __attribute__((ext_vector_type(8))) float v8f;

#ifndef __has_builtin
#define __has_builtin(x) 0
#endif
#if __has_builtin(__builtin_amdgcn_global_load_async_to_lds_b128) && \
    __has_builtin(__builtin_amdgcn_s_wait_asynccnt)
#define USE_ASYNC_LDS 1
#else
#define USE_ASYNC_LDS 0
#endif

#if USE_ASYNC_LDS
// Builtin signature (from clang diagnostic): param0 is int4* in addrspace(1).
// generic->global: flat value == global value. generic->LDS: flat addr low 32 bits
// are the LDS byte offset (ISA §10.2 aperture rules), so integer truncation is safe.
typedef int v4i_vs __attribute__((vector_size(16)));
#define TO_GLOBAL(p) ((__attribute__((address_space(1))) v4i_vs*)(unsigned long long)(size_t)(p))
#define TO_LDS(p)    ((__attribute__((address_space(3))) v4i_vs*)(unsigned int)(size_t)(p))
#endif

__device__ __forceinline__ unsigned rotl32(unsigned v, int r) {
  return (v << r) | (v >> (32 - r));
}

// ---- threefry2x32, key = (0, 42)  (jax.random.key(42)) ----
__device__ __forceinline__ void threefry2x32_42(unsigned& x0, unsigned& x1) {
  const unsigned k0 = 0u, k1 = 42u, k2 = 0u ^ 42u ^ 0x1BD11BDAu;
  x0 += k0; x1 += k1;
  x0 += x1; x1 = rotl32(x1, 13); x1 ^= x0;
  x0 += x1; x1 = rotl32(x1, 15); x1 ^= x0;
  x0 += x1; x1 = rotl32(x1, 26); x1 ^= x0;
  x0 += x1; x1 = rotl32(x1,  6); x1 ^= x0;
  x0 += k1; x1 += k2 + 1u;
  x0 += x1; x1 = rotl32(x1, 17); x1 ^= x0;
  x0 += x1; x1 = rotl32(x1, 29); x1 ^= x0;
  x0 += x1; x1 = rotl32(x1, 16); x1 ^= x0;
  x0 += x1; x1 = rotl32(x1, 24); x1 ^= x0;
  x0 += k2; x1 += k0 + 2u;
  x0 += x1; x1 = rotl32(x1, 13); x1 ^= x0;
  x0 += x1; x1 = rotl32(x1, 15); x1 ^= x0;
  x0 += x1; x1 = rotl32(x1, 26); x1 ^= x0;
  x0 += x1; x1 = rotl32(x1,  6); x1 ^= x0;
  x0 += k0; x1 += k1 + 3u;
  x0 += x1; x1 = rotl32(x1, 17); x1 ^= x0;
  x0 += x1; x1 = rotl32(x1, 29); x1 ^= x0;
  x0 += x1; x1 = rotl32(x1, 16); x1 ^= x0;
  x0 += x1; x1 = rotl32(x1, 24); x1 ^= x0;
  x0 += k1; x1 += k2 + 4u;
  x0 += x1; x1 = rotl32(x1, 13); x1 ^= x0;
  x0 += x1; x1 = rotl32(x1, 15); x1 ^= x0;
  x0 += x1; x1 = rotl32(x1, 26); x1 ^= x0;
  x0 += x1; x1 = rotl32(x1,  6); x1 ^= x0;
  x0 += k2; x1 += k0 + 5u;
}

// grid-seed coordinates from flat index (i5 fastest, matching meshgrid 'ij')
__device__ __forceinline__ void seed_coords(int s, const float* __restrict__ lin, float* c) {
  int r = s;
  #pragma unroll
  for (int d = 5; d >= 0; --d) { int dig = r % 5; r /= 5; c[d] = lin[d] + (float)dig * lin[6 + d]; }
}

// ---------------- K1: sample indices, sigmoid, gather ----------------
__global__ void k_sample(const float* __restrict__ logit, const float* __restrict__ emb,
                         int* __restrict__ idx, float* __restrict__ sp,
                         float* __restrict__ se, float* __restrict__ out_sp) {
  int i = blockIdx.x * blockDim.x + threadIdx.x;
  if (i >= 500) return;
  unsigned x0 = (unsigned)i, x1 = (unsigned)(i + 500);
  threefry2x32_42(x0, x1);
  int i0 = (int)(x0 & 0x1FFFFu);   // span = 131072 = 2^17
  int i1 = (int)(x1 & 0x1FFFFu);
  idx[i] = i0; idx[i + 500] = i1;
  float p0 = 1.0f / (1.0f + __expf(-logit[i0]));
  float p1 = 1.0f / (1.0f + __expf(-logit[i1]));
  sp[i] = p0; sp[i + 500] = p1;
  out_sp[i] = p0; out_sp[i + 500] = p1;
  #pragma unroll
  for (int d = 0; d < 6; ++d) {
    se[i * 6 + d]         = emb[(size_t)i0 * 6 + d];
    se[(i + 500) * 6 + d] = emb[(size_t)i1 * 6 + d];
  }
}

// ---------------- K2: per-dim min/max -> linspace start/step ----------------
__global__ void k_minmax(const float* __restrict__ se, float* __restrict__ lin) {
  __shared__ float smn[256 * 6];
  __shared__ float smx[256 * 6];
  int t = threadIdx.x;
  float mn[6], mx[6];
  #pragma unroll
  for (int d = 0; d < 6; ++d) { mn[d] = __builtin_inff(); mx[d] = -__builtin_inff(); }
  for (int j = t; j < NSAMP; j += 256) {
    #pragma unroll
    for (int d = 0; d < 6; ++d) {
      float v = se[j * 6 + d];
      mn[d] = fminf(mn[d], v); mx[d] = fmaxf(mx[d], v);
    }
  }
  #pragma unroll
  for (int d = 0; d < 6; ++d) { smn[t * 6 + d] = mn[d]; smx[t * 6 + d] = mx[d]; }
  __syncthreads();
  for (int s = 128; s > 0; s >>= 1) {
    if (t < s) {
      #pragma unroll
      for (int d = 0; d < 6; ++d) {
        smn[t * 6 + d] = fminf(smn[t * 6 + d], smn[(t + s) * 6 + d]);
        smx[t * 6 + d] = fmaxf(smx[t * 6 + d], smx[(t + s) * 6 + d]);
      }
    }
    __syncthreads();
  }
  if (t == 0) {
    #pragma unroll
    for (int d = 0; d < 6; ++d) {
      float a = smn[d], b = smx[d];
      float start = a + (b - a) / 12.0f;
      float end   = b - (b - a) / 12.0f;
      lin[d]     = start;
      lin[6 + d] = (end - start) / 4.0f;   // linspace(start,end,5) step
    }
  }
}

// ---------------- K3: seed neighbor-count, filter #1 key ----------------
__global__ void k_seedcount(const float* __restrict__ se, const float* __restrict__ sp,
                            const float* __restrict__ lin, float* __restrict__ key0) {
  __shared__ float s_se[NSAMP * 6];
  __shared__ float s_sp[NSAMP];
  int tid = threadIdx.x;
  for (int k = tid; k < NSAMP * 6; k += 256) s_se[k] = se[k];
  for (int k = tid; k < NSAMP; k += 256) s_sp[k] = sp[k];
  __syncthreads();
  int s = blockIdx.x * 256 + tid;
  if (s >= NSEEDP) return;
  if (s >= NSEED) { key0[s] = -__builtin_inff(); return; }
  float c[6];
  seed_coords(s, lin, c);
  float cnt = 0.0f;
  for (int j = 0; j < NSAMP; ++j) {
    float dsq = 0.0f;
    #pragma unroll
    for (int d = 0; d < 6; ++d) { float df = c[d] - s_se[j * 6 + d] + EPSP; dsq += df * df; }
    if (dsq < BW2) cnt += s_sp[j];
  }
  key0[s] = (cnt > 3.0f) ? cnt : -__builtin_inff();
}

// ---------------- K4: rank-select top-2048 (stable, matches lax.top_k) ----------------
__global__ void k_select1(const float* __restrict__ key0, const float* __restrict__ lin,
                          float* __restrict__ seeds, float* __restrict__ vkey) {
  __shared__ float s_k[256];
  int s = blockIdx.x * 256 + threadIdx.x;
  float my = key0[s];
  int rank = 0;
  for (int base = 0; base < NSEEDP; base += 256) {
    __syncthreads();
    s_k[threadIdx.x] = key0[base + threadIdx.x];
    __syncthreads();
    #pragma unroll 8
    for (int u = 0; u < 256; ++u) {
      int t = base + u; float kt = s_k[u];
      rank += ((kt > my) || (kt == my && t < s)) ? 1 : 0;
    }
  }
  if (rank < SMAX) {
    int sd = (s < NSEED) ? s : (NSEED - 1);
    float c[6];
    seed_coords(sd, lin, c);
    #pragma unroll
    for (int d = 0; d < 6; ++d) seeds[rank * 6 + d] = c[d];
    vkey[rank] = my;
  }
}

// ---------------- K5: 5 mean-shift iterations + filter #2 key ----------------
__global__ void k_meanshift(const float* __restrict__ se, const float* __restrict__ sp,
                            float* __restrict__ seeds, const float* __restrict__ vkey,
                            float* __restrict__ key2) {
  __shared__ float s_se[NSAMP * 6];
  __shared__ float s_sp[NSAMP];
  int tid = threadIdx.x;
  for (int k = tid; k < NSAMP * 6; k += 256) s_se[k] = se[k];
  for (int k = tid; k < NSAMP; k += 256) s_sp[k] = sp[k];
  __syncthreads();
  int s = blockIdx.x * 256 + tid;
  float c[6];
  #pragma unroll
  for (int d = 0; d < 6; ++d) c[d] = seeds[s * 6 + d];
  for (int it = 0; it < 5; ++it) {
    float num[6] = {0, 0, 0, 0, 0, 0};
    float den = 0.0f;
    for (int j = 0; j < NSAMP; ++j) {
      float dsq = 0.0f;
      #pragma unroll
      for (int d = 0; d < 6; ++d) { float df = c[d] - s_se[j * 6 + d] + EPSP; dsq += df * df; }
      float w = __expf(INVH * dsq) * s_sp[j];   // GAUSS factor cancels in normalization
      den += w;
      #pragma unroll
      for (int d = 0; d < 6; ++d) num[d] += w * s_se[j * 6 + d];
    }
    #pragma unroll
    for (int d = 0; d < 6; ++d) c[d] = num[d] / den;
  }
  #pragma unroll
  for (int d = 0; d < 6; ++d) seeds[s * 6 + d] = c[d];
  float cnt = 0.0f;
  for (int j = 0; j < NSAMP; ++j) {
    float dsq = 0.0f;
    #pragma unroll
    for (int d = 0; d < 6; ++d) { float df = c[d] - s_se[j * 6 + d] + EPSP; dsq += df * df; }
    if (dsq < BW2) cnt += s_sp[j];
  }
  bool valid = (vkey[s] != -__builtin_inff());
  key2[s] = (valid && cnt > 10.0f) ? cnt : -__builtin_inff();
}

// ---------------- K6: rank-select top-256 ----------------
__global__ void k_select2(const float* __restrict__ key2, const float* __restrict__ seeds,
                          float* __restrict__ mseed, float* __restrict__ mkey) {
  __shared__ float s_k[SMAX];
  int tid = threadIdx.x;
  for (int k = tid; k < SMAX; k += 256) s_k[k] = key2[k];
  __syncthreads();
  int s = blockIdx.x * 256 + tid;
  float my = key2[s];
  int rank = 0;
  #pragma unroll 8
  for (int t = 0; t < SMAX; ++t) {
    float kt = s_k[t];
    rank += ((kt > my) || (kt == my && t < s)) ? 1 : 0;
  }
  if (rank < KMAX) {
    #pragma unroll
    for (int d = 0; d < 6; ++d) mseed[rank * 6 + d] = seeds[s * 6 + d];
    mkey[rank] = my;
  }
}

// ---------------- K7: greedy NMS merge (serial scan in one WG) ----------------
__global__ void k_merge(const float* __restrict__ mseed, const float* __restrict__ mkey,
                        float* __restrict__ ctr, float* __restrict__ csum,
                        float* __restrict__ cn2, float* __restrict__ cval,
                        float* __restrict__ out_ctr) {
  __shared__ float S[KMAX * 6];
  __shared__ float s_valid[KMAX];
  __shared__ float ikey[KMAX];
  __shared__ int   order[KMAX];
  __shared__ int   isc[KMAX];
  __shared__ int   labels_s[KMAX];
  __shared__ int   labels[KMAX];
  __shared__ float vq[KMAX];
  int t = threadIdx.x;
  for (int k = t; k < KMAX * 6; k += 256) S[k] = mseed[k];
  s_valid[t] = (mkey[t] != -__builtin_inff()) ? 1.0f : 0.0f;
  __syncthreads();
  // intensity
  float inten = 0.0f;
  for (int j = 0; j < KMAX; ++j) {
    float dsq = 0.0f;
    #pragma unroll
    for (int d = 0; d < 6; ++d) { float df = S[t * 6 + d] - S[j * 6 + d] + EPSP; dsq += df * df; }
    if (dsq < BW2) inten += s_valid[j];
  }
  float key = (s_valid[t] > 0.0f) ? inten : -__builtin_inff();
  ikey[t] = key;
  __syncthreads();
  // argsort(x)[::-1]: descending, ties -> larger index first
  int rank = 0;
  for (int j = 0; j < KMAX; ++j) {
    float kj = ikey[j];
    rank += ((kj > key) || (kj == key && j > t)) ? 1 : 0;
  }
  order[rank] = t;
  isc[t] = 1;
  labels_s[t] = -1;
  __syncthreads();
  vq[t] = s_valid[order[t]];
  int cur = 0;
  for (int i = 0; i < KMAX; ++i) {
    __syncthreads();
    int active = (isc[i] != 0) && (vq[i] > 0.0f);
    if (active) {
      if (t == i) {
        labels_s[t] = cur;
      } else if (t > i && isc[t] != 0 && vq[t] > 0.0f) {
        int oi = order[i], ot = order[t];
        float dsq = 0.0f;
        #pragma unroll
        for (int d = 0; d < 6; ++d) { float df = S[oi * 6 + d] - S[ot * 6 + d] + EPSP; dsq += df * df; }
        if (dsq < BW2) { labels_s[t] = cur; isc[t] = 0; }
      }
    }
    cur += active ? 1 : 0;
  }
  __syncthreads();
  labels[order[t]] = labels_s[t];
  __syncthreads();
  // cluster means
  float acc[6] = {0, 0, 0, 0, 0, 0};
  float cnt = 0.0f;
  for (int r = 0; r < KMAX; ++r) {
    if (labels[r] == t && s_valid[r] > 0.0f) {
      cnt += 1.0f;
      #pragma unroll
      for (int d = 0; d < 6; ++d) acc[d] += S[r * 6 + d];
    }
  }
  float cc[6];
  float s1 = 0.0f, s2 = 0.0f;
  #pragma unroll
  for (int d = 0; d < 6; ++d) {
    cc[d] = (cnt > 0.0f) ? (acc[d] / cnt) : 0.0f;
    s1 += cc[d]; s2 += cc[d] * cc[d];
    ctr[t * 6 + d] = cc[d];
    out_ctr[t * 6 + d] = cc[d];
  }
  csum[t] = s1; cn2[t] = s2;
  cval[t] = (cnt > 0.0f) ? 1.0f : 0.0f;
}

// ---------------- K8: WMMA clustering + softmax (the bandwidth kernel) ----------------
__global__ void __launch_bounds__(256, 1)
k_cluster(const float* __restrict__ emb, const float* __restrict__ ctr,
          const float* __restrict__ csum, const float* __restrict__ cn2,
          const float* __restrict__ cval, float* __restrict__ seg,
          float* __restrict__ outinv) {
  __shared__ float sRaw[KMAX * 6];     // raw centers (async-copied)
  __shared__ float sA0[16 * 32 * 2];   // A fragments, K=0..3 (pre-swizzled, b64 per lane)
  __shared__ float sA1[16 * 32 * 2];   // A fragments, K=4..7 (zero-padded)
  __shared__ float sQ[KMAX];           // |c|^2 - 2*eps*sum(c)
  __shared__ float sVf[KMAX];          // validity mask as float
  int tid = threadIdx.x;

#if USE_ASYNC_LDS
  for (int k = tid; k < (KMAX * 6) / 4; k += 256)
    __builtin_amdgcn_global_load_async_to_lds_b128(TO_GLOBAL(ctr + 4 * k),
                                                   TO_LDS(sRaw + 4 * k), 0, 0);
  __builtin_amdgcn_s_wait_asynccnt(0);
#else
  for (int k = tid; k < KMAX * 6; k += 256) sRaw[k] = ctr[k];
#endif
  sQ[tid]  = cn2[tid] - 2.0f * EPSP * csum[tid];
  sVf[tid] = cval[tid];
  __syncthreads();

  // Build swizzled A fragments from raw centers (unconditional loads; zero-pad by multiply).
  for (int s = tid; s < 512; s += 256) {
    int t = s >> 5, ln = s & 31;
    int m = t * 16 + (ln & 15);
    int o = (ln < 16) ? 0 : 2;
    float msk = (ln < 16) ? 1.0f : 0.0f;
    float r0 = sRaw[m * 6 + o];
    float r1 = sRaw[m * 6 + o + 1];
    float r4 = sRaw[m * 6 + 4];
    float r5 = sRaw[m * 6 + 5];
    sA0[s * 2 + 0] = r0;
    sA0[s * 2 + 1] = r1;
    sA1[s * 2 + 0] = msk * r4;
    sA1[s * 2 + 1] = msk * r5;
  }
  __syncthreads();

  const int lane = tid & 31;
  const int wv   = tid >> 5;
  const bool hi  = (lane >= 16);
  const int prow = (blockIdx.x << 7) + (wv << 4) + (lane & 15);   // point index

  const float* ep = emb + (size_t)prow * 6;
  float2 p01 = *(const float2*)(ep);       // rows are 24B, 8B-aligned
  float2 p23 = *(const float2*)(ep + 2);
  float2 p45 = *(const float2*)(ep + 4);
  if (blockIdx.x + 1 < gridDim.x)
    __builtin_prefetch(ep + 128 * 6, 0, 1);   // global_prefetch_b8

  // B fragments (points are the N dimension): K=0..3 then K=4..7 (dims 4,5 + pad)
  v2f b0; b0[0] = hi ? p23.x : p01.x;  b0[1] = hi ? p23.y : p01.y;
  v2f b1; b1[0] = hi ? 0.0f : p45.x;   b1[1] = hi ? 0.0f : p45.y;

  const float en2 = p01.x*p01.x + p01.y*p01.y + p23.x*p23.x + p23.y*p23.y +
                    p45.x*p45.x + p45.y*p45.y;
  const float es  = p01.x + p01.y + p23.x + p23.y + p45.x + p45.y;
  const float ebase = en2 + 2.0f * EPSP * es + 6.0f * EPSP * EPSP;

  v8f acc[16];
  #pragma unroll
  for (int t = 0; t < 16; ++t) {
    v2f a0 = *(const v2f*)&sA0[(t * 32 + lane) * 2];   // ds_load_b64
    v2f a1 = *(const v2f*)&sA1[(t * 32 + lane) * 2];   // ds_load_b64
    v8f z = {0.0f, 0.0f, 0.0f, 0.0f, 0.0f, 0.0f, 0.0f, 0.0f};
    z = __builtin_amdgcn_wmma_f32_16x16x4_f32(false, a0, false, b0, (short)0, z, false, false);
    z = __builtin_amdgcn_wmma_f32_16x16x4_f32(false, a1, false, b1, (short)0, z, false, false);
    acc[t] = z;
  }

  const int mo = hi ? 8 : 0;
  float maxv = -__builtin_inff();
  #pragma unroll
  for (int t = 0; t < 16; ++t) {
    #pragma unroll
    for (int r = 0; r < 8; ++r) {
      int ci = t * 16 + r + mo;
      float dsq = fmaxf(ebase + sQ[ci] - 2.0f * acc[t][r], 0.0f);
      float sc = 1.0f / (sqrtf(dsq) + 0.01f);
      maxv = fmaxf(maxv, (sVf[ci] > 0.0f) ? sc : -__builtin_inff());
    }
  }
  maxv = fmaxf(maxv, __shfl_xor(maxv, 16, 32));
  if (!hi) outinv[prow] = (maxv < 7.0f) ? 1.0f : 0.0f;

  float ssum = 0.0f;
  #pragma unroll
  for (int t = 0; t < 16; ++t) {
    #pragma unroll
    for (int r = 0; r < 8; ++r) {
      int ci = t * 16 + r + mo;
      float dsq = fmaxf(ebase + sQ[ci] - 2.0f * acc[t][r], 0.0f);
      float sc = 1.0f / (sqrtf(dsq) + 0.01f);
      float ev = (sVf[ci] > 0.0f) ? __expf(sc - maxv) : 0.0f;
      acc[t][r] = ev;
      ssum += ev;
    }
  }
  ssum += __shfl_xor(ssum, 16, 32);
  const float inv = 1.0f / ssum;

  #pragma unroll
  for (int t = 0; t < 16; ++t) {
    float4 lo4 = make_float4(acc[t][0] * inv, acc[t][1] * inv, acc[t][2] * inv, acc[t][3] * inv);
    float4 hi4 = make_float4(acc[t][4] * inv, acc[t][5] * inv, acc[t][6] * inv, acc[t][7] * inv);
    float* base = seg + (size_t)prow * KMAX + t * 16 + mo;
    *(float4*)(base)     = lo4;   // global_store_b128
    *(float4*)(base + 4) = hi4;
  }
}

// ---------------- launcher ----------------
extern "C" void kernel_launch(void* const* d_in, const int* in_sizes, int n_in,
                              void* d_out, int out_size, void* d_ws, size_t ws_size,
                              hipStream_t stream) {
  const float* logit = (const float*)d_in[0];
  const float* emb   = (const float*)d_in[1];
  float* out = (float*)d_out;

  float* out_seg = out;                                  // NPTS*KMAX
  float* out_ctr = out + (size_t)NPTS * KMAX;            // KMAX*6
  float* out_inv = out_ctr + KMAX * 6;                   // NPTS
  float* out_sp  = out_inv + NPTS;                       // NSAMP

  float* w = (float*)d_ws;
  int*   idx   = (int*)w;          // 1000 ints
  float* sp    = w + 1024;         // 1000
  float* se    = w + 2048;         // 6000
  float* lin   = w + 8192;         // 12 (start[6], step[6])
  float* key0  = w + 8320;         // 16384
  float* seeds = w + 24704;        // 2048*6
  float* vkey  = w + 37120;        // 2048
  float* key2  = w + 39168;        // 2048
  float* mseed = w + 41216;        // 256*6
  float* mkey  = w + 42752;        // 256
  float* ctr   = w + 43008;        // 256*6
  float* csum  = w + 44544;        // 256
  float* cn2   = w + 44800;        // 256
  float* cval  = w + 45056;        // 256  (total ~177 KB)

  k_sample   <<<2,            256, 0, stream>>>(logit, emb, idx, sp, se, out_sp);
  k_minmax   <<<1,            256, 0, stream>>>(se, lin);
  k_seedcount<<<NSEEDP / 256, 256, 0, stream>>>(se, sp, lin, key0);
  k_select1  <<<NSEEDP / 256, 256, 0, stream>>>(key0, lin, seeds, vkey);
  k_meanshift<<<SMAX / 256,   256, 0, stream>>>(se, sp, seeds, vkey, key2);
  k_select2  <<<SMAX / 256,   256, 0, stream>>>(key2, seeds, mseed, mkey);
  k_merge    <<<1,            256, 0, stream>>>(mseed, mkey, ctr, csum, cn2, cval, out_ctr);
  k_cluster  <<<NPTS / 128,   256, 0, stream>>>(emb, ctr, csum, cn2, cval, out_seg, out_inv);
}